// UnfusedGemma4TextExperts_75402445849172
// MI455X (gfx1250) — compile-verified
//
#include <hip/hip_runtime.h>
#include <math.h>

// ---------------------------------------------------------------------------
// Dense MoE (all experts process all tokens), fp32, CDNA5 WMMA f32 16x16x4.
//   hidden_states [32,2048], gate_w/up_w [8,1024,2048], down_w [8,2048,1024]
//   out [32,2048] fp32 = mean_e( (gelu_tanh(hs.gwT) * (hs.uwT)) . dwT )
//
// Roofline: 3.2 GFLOP / 201 MB fp32 weights -> 16 FLOP/B, memory bound,
// ~8.6 us floor @ 23.3 TB/s.  Weights are streamed ONCE and are larger than
// the 192 MB L2, so weight loads use non-temporal hints (TH=NT) to keep the
// reused working set (hs: 256 KB, act: 1 MB, read ~128x) resident in L2.
// ---------------------------------------------------------------------------

typedef __attribute__((ext_vector_type(2))) float v2f;
typedef __attribute__((ext_vector_type(8))) float v8f;

static constexpr int kT = 32;    // tokens
static constexpr int kH = 2048;  // hidden
static constexpr int kI = 1024;  // intermediate
static constexpr int kE = 8;     // experts

__device__ __forceinline__ float gelu_tanh(float x) {
  // jax.nn.gelu(approximate=True)
  const float c0 = 0.7978845608028654f;  // sqrt(2/pi)
  const float c1 = 0.044715f;
  return 0.5f * x * (1.0f + tanhf(c0 * (x + c1 * x * x * x)));
}

__device__ __forceinline__ v2f ld_stream(const float* p) {
  // Non-temporal: streamed-once weight data, evict early from GL2.
  return __builtin_nontemporal_load((const v2f*)p);
}

// ---------------------------------------------------------------------------
// Kernel 1: act[e][t][i] = gelu(hs . gate_w[e]^T) * (hs . up_w[e]^T)
// One wave per (expert, 16-wide I tile): GEMM M=32 (two 16-tiles), N=16, K=2048.
// Per k-step: 2 A frags (m0,m1), 2 B frags (gate,up) -> 4 WMMAs; B reused 2x.
// Fragment layout (32-bit operands, ISA 7.12.2):
//   A 16x4: lane m = lane&15, VGPR pair holds K = {2*(lane>>4), 2*(lane>>4)+1}
//   B 4x16: lane n = lane&15, same K-pair selection (transpose-symmetric)
//   C 16x16: VGPR r -> M = r + 8*(lane>>4), N = lane&15
// ---------------------------------------------------------------------------
__global__ __launch_bounds__(32) void moe_gateup_wmma(
    const float* __restrict__ hs, const float* __restrict__ gw,
    const float* __restrict__ uw, float* __restrict__ act) {
  const int e    = blockIdx.x >> 6;   // 64 I-tiles per expert
  const int nt   = blockIdx.x & 63;
  const int lane = threadIdx.x;
  const int half = lane >> 4;
  const int l16  = lane & 15;
  const int n    = nt * 16 + l16;     // I column
  const int koff = 2 * half;

  const float* __restrict__ gwe = gw + ((size_t)e * kI + n) * kH;
  const float* __restrict__ uwe = uw + ((size_t)e * kI + n) * kH;
  const float* __restrict__ hs0 = hs + (size_t)l16 * kH;         // M tile 0
  const float* __restrict__ hs1 = hs + (size_t)(l16 + 16) * kH;  // M tile 1

  v8f g0 = {}, g1 = {}, u0 = {}, u1 = {};

#pragma unroll 8
  for (int k = 0; k < kH; k += 4) {
    v2f a0 = *(const v2f*)(hs0 + k + koff);      // hs: RT (hot in L2)
    v2f a1 = *(const v2f*)(hs1 + k + koff);
    v2f bg = ld_stream(gwe + k + koff);          // weights: NT stream
    v2f bu = ld_stream(uwe + k + koff);
    g0 = __builtin_amdgcn_wmma_f32_16x16x4_f32(false, a0, false, bg, (short)0, g0, false, false);
    g1 = __builtin_amdgcn_wmma_f32_16x16x4_f32(false, a1, false, bg, (short)0, g1, false, false);
    u0 = __builtin_amdgcn_wmma_f32_16x16x4_f32(false, a0, false, bu, (short)0, u0, false, false);
    u1 = __builtin_amdgcn_wmma_f32_16x16x4_f32(false, a1, false, bu, (short)0, u1, false, false);
  }

  float* __restrict__ actp = act + (size_t)e * kT * kI;
#pragma unroll
  for (int r = 0; r < 8; ++r) {
    const int m0 = r + 8 * half;
    actp[(size_t)m0 * kI + n]        = gelu_tanh(g0[r]) * u0[r];
    actp[(size_t)(m0 + 16) * kI + n] = gelu_tanh(g1[r]) * u1[r];
  }
}

// ---------------------------------------------------------------------------
// Kernel 2: partial[e][t][h] = act[e] . down_w[e]^T   (M=32, N=2048, K=1024)
// Split-K over experts for parallelism (1024 waves); summed in kernel 3 to
// keep the reduction deterministic (no float atomics).
// ---------------------------------------------------------------------------
__global__ __launch_bounds__(32) void moe_down_wmma(
    const float* __restrict__ act, const float* __restrict__ dw,
    float* __restrict__ partial) {
  const int e    = blockIdx.x >> 7;   // 128 H-tiles per expert
  const int nt   = blockIdx.x & 127;
  const int lane = threadIdx.x;
  const int half = lane >> 4;
  const int l16  = lane & 15;
  const int n    = nt * 16 + l16;     // H column
  const int koff = 2 * half;

  const float* __restrict__ dwe = dw  + ((size_t)e * kH + n)        * kI;
  const float* __restrict__ a0p = act + ((size_t)e * kT + l16)      * kI;
  const float* __restrict__ a1p = act + ((size_t)e * kT + l16 + 16) * kI;

  v8f c0 = {}, c1 = {};

#pragma unroll 8
  for (int k = 0; k < kI; k += 4) {
    v2f a0 = *(const v2f*)(a0p + k + koff);      // act: RT (L2-resident, read 128x)
    v2f a1 = *(const v2f*)(a1p + k + koff);
    v2f b  = ld_stream(dwe + k + koff);          // weights: NT stream
    c0 = __builtin_amdgcn_wmma_f32_16x16x4_f32(false, a0, false, b, (short)0, c0, false, false);
    c1 = __builtin_amdgcn_wmma_f32_16x16x4_f32(false, a1, false, b, (short)0, c1, false, false);
  }

  float* __restrict__ pp = partial + (size_t)e * kT * kH;
#pragma unroll
  for (int r = 0; r < 8; ++r) {
    const int m0 = r + 8 * half;
    pp[(size_t)m0 * kH + n]        = c0[r];
    pp[(size_t)(m0 + 16) * kH + n] = c1[r];
  }
}

// ---------------------------------------------------------------------------
// Kernel 3: out[t][h] = (1/8) * sum_e partial[e][t][h]
// ---------------------------------------------------------------------------
__global__ __launch_bounds__(256) void moe_mean(const float* __restrict__ partial,
                                                float* __restrict__ out) {
  const int idx = blockIdx.x * 256 + threadIdx.x;
  if (idx < kT * kH) {
    float s = 0.0f;
#pragma unroll
    for (int e = 0; e < kE; ++e) s += partial[(size_t)e * kT * kH + idx];
    out[idx] = s * 0.125f;
  }
}

extern "C" void kernel_launch(void* const* d_in, const int* in_sizes, int n_in,
                              void* d_out, int out_size, void* d_ws, size_t ws_size,
                              hipStream_t stream) {
  const float* hs = (const float*)d_in[0];  // [32,2048]
  // d_in[1] routing_weights, d_in[2] selected_experts: unused by reference
  const float* gw = (const float*)d_in[3];  // [8,1024,2048]
  const float* uw = (const float*)d_in[4];  // [8,1024,2048]
  const float* dw = (const float*)d_in[5];  // [8,2048,1024]
  float* out = (float*)d_out;               // [32,2048]

  // Workspace layout: act [8,32,1024] (1 MB) | partial [8,32,2048] (2 MB)
  float* act     = (float*)d_ws;
  float* partial = act + (size_t)kE * kT * kI;

  moe_gateup_wmma<<<kE * (kI / 16), 32, 0, stream>>>(hs, gw, uw, act);
  moe_down_wmma<<<kE * (kH / 16), 32, 0, stream>>>(act, dw, partial);
  moe_mean<<<(kT * kH + 255) / 256, 256, 0, stream>>>(partial, out);
}